// HOGLayerTorch_64467459113398
// MI455X (gfx1250) — compile-verified
//
#include <hip/hip_runtime.h>
#include <math.h>

#define NBINS   9
#define IMG_H   512
#define IMG_W   512
#define CHN     3
#define STRIPS  64            // IMG_H / 8
#define ROWS    10            // 8 interior rows + 2 halo rows
#define GW      514           // gray buffer row stride: 512 + 2 halo cols
#define NIMG    32

typedef unsigned int u32x4 __attribute__((ext_vector_type(4)));
typedef int          i32x8 __attribute__((ext_vector_type(8)));
typedef int          i32x4 __attribute__((ext_vector_type(4)));

__global__ __launch_bounds__(256)
void hog_fused_kernel(const float* __restrict__ x,
                      const float* __restrict__ wgt,
                      float* __restrict__ out)
{
    // LDS: TDM-staged channel strip, gray strip, bin histogram, weights
    __shared__ __attribute__((aligned(16))) float ch[CHN * ROWS * IMG_W]; // 61440 B
    __shared__ float gray[ROWS * GW];          // 20560 B
    __shared__ float hist[NBINS * 64];         //  2304 B
    __shared__ float wsh[18];                  //    72 B

    const int tid   = threadIdx.x;
    const int blk   = blockIdx.x;
    const int n     = blk >> 6;        // image index
    const int strip = blk & 63;        // pooled row index
    const int y0    = strip * 8;

    // ---- Tensor Data Mover: stage 3 channels x (<=10) rows x 512 f32 -> LDS.
    // One wave issues the DMAs (TDM ignores EXEC; issue exactly once per block).
    // Bottom border: tensor_dim1 OOB rows are zero-filled by TDM.
    // Top border: skip row 0 of the LDS strip (gray pass zero-fills it).
    if (tid == 0) {
        const int      rskip  = (strip == 0) ? 1 : 0;
        const int      ystart = y0 - 1 + rskip;
        const unsigned rows   = (unsigned)(ROWS - rskip);      // tile_dim1
        const unsigned tdim1  = (unsigned)(IMG_H - ystart);    // rows to tensor end
        #pragma unroll
        for (int c = 0; c < CHN; ++c) {
            unsigned long long ga = (unsigned long long)(size_t)
                (x + (((size_t)(n * CHN + c)) * IMG_H + ystart) * IMG_W);
            unsigned lds = (unsigned)(size_t)&ch[(c * ROWS + rskip) * IMG_W];

            u32x4 g0;
            g0.x = 1u;                                   // count=1, user D#, no gather
            g0.y = lds;                                  // lds_addr (bytes)
            g0.z = (unsigned)ga;                         // global_addr[31:0]
            g0.w = (unsigned)((ga >> 32) & 0x01FFFFFFu)  // global_addr[56:32]
                 | (2u << 30);                           // type = 2 ("image")

            i32x8 g1;
            g1[0] = (int)(2u << 16);                     // data_size = 2 -> 4 bytes
            g1[1] = (int)((unsigned)IMG_W << 16);        // tensor_dim0 = 512
            g1[2] = (int)((tdim1 & 0xFFFFu) << 16);      // tensor_dim1 (zero-fill past end)
            g1[3] = (int)((unsigned)IMG_W << 16);        // tile_dim0 = 512
            g1[4] = (int)rows;                           // tile_dim1 (tile_dim2 = 0)
            g1[5] = (int)IMG_W;                          // tensor_dim0_stride = 512
            g1[6] = 0;                                   // stride hi / dim1_stride unused (2D)
            g1[7] = 0;

            i32x4 gz4 = {0, 0, 0, 0};                    // groups 2/3: unused dims
            i32x8 gz8 = {0, 0, 0, 0, 0, 0, 0, 0};        // extra descriptor words (unused)
            __builtin_amdgcn_tensor_load_to_lds(g0, g1, gz4, gz4, gz8, 0);
        }
        __builtin_amdgcn_s_wait_tensorcnt(0);
    }

    // Overlaps with the DMA: zero histogram, stage conv weights.
    for (int i = tid; i < NBINS * 64; i += 256) hist[i] = 0.0f;
    if (tid < 18) wsh[tid] = wgt[tid];

    __syncthreads();

    // ---- Gray pass: mean over channels, zero-pad out-of-image rows/cols -----
    const float inv3 = 1.0f / 3.0f;
    #pragma unroll
    for (int i = 0; i < 5; ++i) {
        int e4 = tid + i * 256;          // 1280 groups of 4 floats (10*512 total)
        int r  = e4 >> 7;                // row within strip
        int x4 = (e4 & 127) * 4;         // column (16B aligned)
        int y  = y0 - 1 + r;
        float g0v = 0.0f, g1v = 0.0f, g2v = 0.0f, g3v = 0.0f;
        if (y >= 0 && y < IMG_H) {
            const float4 a = *(const float4*)&ch[(0 * ROWS + r) * IMG_W + x4];
            const float4 b = *(const float4*)&ch[(1 * ROWS + r) * IMG_W + x4];
            const float4 c = *(const float4*)&ch[(2 * ROWS + r) * IMG_W + x4];
            g0v = (a.x + b.x + c.x) * inv3;
            g1v = (a.y + b.y + c.y) * inv3;
            g2v = (a.z + b.z + c.z) * inv3;
            g3v = (a.w + b.w + c.w) * inv3;
        }
        int gb = r * GW + x4 + 1;        // +1: left halo column
        gray[gb + 0] = g0v;
        gray[gb + 1] = g1v;
        gray[gb + 2] = g2v;
        gray[gb + 3] = g3v;
    }
    if (tid < 2 * ROWS) {                // zero halo columns (conv zero-pad)
        int r = tid >> 1;
        gray[r * GW + ((tid & 1) ? (GW - 1) : 0)] = 0.0f;
    }
    __syncthreads();

    // ---- Gradient, phase binning, pooled accumulation -----------------------
    const float binscale = (float)NBINS / (float)M_PI;
    #pragma unroll 4
    for (int i = 0; i < 16; ++i) {
        int p  = tid + i * 256;          // 8*512 = 4096 pixels
        int r  = p >> 9;                 // interior row 0..7
        int xx = p & 511;                // image column
        int gr = r + 1;                  // gray-buffer row

        float gxv = 0.0f, gyv = 0.0f;
        #pragma unroll
        for (int dy = 0; dy < 3; ++dy) {
            int base = (gr - 1 + dy) * GW + xx;   // buffer col = xx + dx (halo shift)
            #pragma unroll
            for (int dx = 0; dx < 3; ++dx) {
                float g = gray[base + dx];
                gxv = fmaf(wsh[dy * 3 + dx], g, gxv);
                gyv = fmaf(wsh[9 + dy * 3 + dx], g, gyv);
            }
        }

        float nrm = sqrtf(gxv * gxv + gyv * gyv);
        float pb  = atan2f(gxv, gyv) * binscale;   // (-9, 9]
        int lo = (int)floorf(pb);
        int hi = (int)ceilf(pb);
        lo %= NBINS; if (lo < 0) lo += NBINS;      // floored mod, like jnp %
        hi %= NBINS; if (hi < 0) hi += NBINS;

        int pc = xx >> 3;                          // pooled column
        atomicAdd(&hist[lo * 64 + pc], nrm);
        atomicAdd(&hist[hi * 64 + pc], 1.0f - nrm);
    }
    __syncthreads();

    // ---- Pooled mean (1/64) + coalesced store -------------------------------
    const float inv64 = 1.0f / 64.0f;
    for (int i = tid; i < NBINS * 64; i += 256) {
        int bin = i >> 6;
        int pc  = i & 63;
        out[(((size_t)n * NBINS + bin) * STRIPS + strip) * 64 + pc] =
            hist[i] * inv64;
    }
}

extern "C" void kernel_launch(void* const* d_in, const int* in_sizes, int n_in,
                              void* d_out, int out_size, void* d_ws, size_t ws_size,
                              hipStream_t stream)
{
    const float* x   = (const float*)d_in[0];   // (32,3,512,512) f32
    const float* wgt = (const float*)d_in[1];   // (2,1,3,3) f32
    float* out = (float*)d_out;                 // (32,9,64,64) f32

    hog_fused_kernel<<<dim3(NIMG * STRIPS), dim3(256), 0, stream>>>(x, wgt, out);
}